// SourceGenerator_13434657702538
// MI455X (gfx1250) — compile-verified
//
#include <hip/hip_runtime.h>

typedef __attribute__((ext_vector_type(2))) float v2f;
typedef __attribute__((ext_vector_type(8))) float v8f;

constexpr int B = 64;
constexpr int N = 10000;
constexpr int D = 128;
constexpr int F = 4;
constexpr int KDIM = (F + 1) * D;        // 640
constexpr int CHUNKS = 25;               // per batch element
constexpr int ROWS_PER_CHUNK = N / CHUNKS; // 400

// ---------------------------------------------------------------------------
// K1: partial sums of H over the N dimension.
// grid = B*CHUNKS blocks x 256 threads. Each block reduces 400 rows of 128
// floats with float4 (b128) loads; deterministic fixed-order accumulation.
// ---------------------------------------------------------------------------
__global__ void k_reduce_partial(const float* __restrict__ H,
                                 float* __restrict__ partial) {
    const int blk   = blockIdx.x;
    const int b     = blk / CHUNKS;
    const int chunk = blk % CHUNKS;
    const int tid   = threadIdx.x;
    const int lane  = tid & 31;   // float4 column 0..31  (d = lane*4..lane*4+3)
    const int wrp   = tid >> 5;   // 0..7 row-group

    const float4* __restrict__ Hb =
        (const float4*)(H + (size_t)b * N * D);

    float4 acc = make_float4(0.f, 0.f, 0.f, 0.f);
    const int r0 = chunk * ROWS_PER_CHUNK;
    #pragma unroll 2
    for (int r = r0 + wrp; r < r0 + ROWS_PER_CHUNK; r += 8) {
        float4 v = Hb[(size_t)r * 32 + lane];
        acc.x += v.x; acc.y += v.y; acc.z += v.z; acc.w += v.w;
    }

    __shared__ float lds[8 * 128];
    lds[wrp * 128 + lane * 4 + 0] = acc.x;
    lds[wrp * 128 + lane * 4 + 1] = acc.y;
    lds[wrp * 128 + lane * 4 + 2] = acc.z;
    lds[wrp * 128 + lane * 4 + 3] = acc.w;
    __syncthreads();

    if (tid < 128) {
        float s = 0.f;
        #pragma unroll
        for (int w = 0; w < 8; ++w) s += lds[w * 128 + tid];
        partial[(size_t)blk * 128 + tid] = s;
    }
}

// ---------------------------------------------------------------------------
// K2: fold the 25 partials into the mean row and build emb (64 x 640):
//   emb[b][f*128+d] = relu(H[b, indice[b,f], d])  for f = 0..3
//   emb[b][512 +d]  = mean over N
// grid = 64 blocks x 128 threads.
// ---------------------------------------------------------------------------
__global__ void k_build_emb(const float* __restrict__ H,
                            const int* __restrict__ indice,
                            const float* __restrict__ partial,
                            float* __restrict__ emb) {
    const int b = blockIdx.x;
    const int d = threadIdx.x;

    float s = 0.f;
    #pragma unroll
    for (int c = 0; c < CHUNKS; ++c)
        s += partial[((size_t)(b * CHUNKS + c)) * 128 + d];
    emb[(size_t)b * KDIM + 4 * 128 + d] = s * (1.0f / (float)N);

    #pragma unroll
    for (int f = 0; f < F; ++f) {
        int idx = indice[b * F + f];
        float v = H[((size_t)b * N + idx) * D + d];
        emb[(size_t)b * KDIM + f * 128 + d] = fmaxf(v, 0.0f);
    }
}

// ---------------------------------------------------------------------------
// K3: x = relu(emb @ W + bias); layernorm(x) -> out.
// One workgroup of 1024 threads = 32 waves. Wave w owns C tile
// (tileM = w/8, tileN = w%8); K loop of 160 x V_WMMA_F32_16X16X4_F32.
//   A 16x4 layout: lane -> M = lane%16, K = vgpr + 2*(lane/16)
//   B  4x16 layout: lane -> N = lane%16, K = vgpr + 2*(lane/16)
//   C 16x16 layout: vgpr r, lane -> row = r + 8*(lane/16), col = lane%16
// ---------------------------------------------------------------------------
__global__ void k_gemm_ln(const float* __restrict__ emb,
                          const float* __restrict__ W,
                          const float* __restrict__ bias,
                          const float* __restrict__ gamma,
                          const float* __restrict__ beta,
                          float* __restrict__ out) {
    __shared__ float xbuf[64 * 132];     // padded stride 132 (34 KB)

    const int tid   = threadIdx.x;
    const int wave  = tid >> 5;          // 0..31
    const int lane  = tid & 31;
    const int tileM = wave >> 3;         // 0..3
    const int tileN = wave & 7;          // 0..7

    const int m    = tileM * 16 + (lane & 15);
    const int n    = tileN * 16 + (lane & 15);
    const int koff = (lane >> 4) * 2;    // 0 or 2

    const float* __restrict__ Arow = emb + (size_t)m * KDIM;

    v8f c = {};
    for (int k = 0; k < KDIM; k += 4) {
        v2f a, bm;
        a.x  = Arow[k + koff];
        a.y  = Arow[k + koff + 1];
        bm.x = W[(size_t)(k + koff)     * D + n];
        bm.y = W[(size_t)(k + koff + 1) * D + n];
        c = __builtin_amdgcn_wmma_f32_16x16x4_f32(
                /*neg_a=*/false, a, /*neg_b=*/false, bm,
                /*c_mod=*/(short)0, c, /*reuse_a=*/false, /*reuse_b=*/false);
    }

    // bias + relu, stage to LDS
    const int mbase = tileM * 16 + 8 * (lane >> 4);
    const int col   = tileN * 16 + (lane & 15);
    const float bv  = bias[col];
    #pragma unroll
    for (int r = 0; r < 8; ++r) {
        float x = fmaxf(c[r] + bv, 0.0f);
        xbuf[(mbase + r) * 132 + col] = x;
    }
    __syncthreads();

    // LayerNorm: each wave handles rows {wave, wave+32}; lane owns 4 cols.
    for (int row = wave; row < 64; row += 32) {
        float vals[4];
        float s = 0.f, sq = 0.f;
        #pragma unroll
        for (int j = 0; j < 4; ++j) {
            float v = xbuf[row * 132 + lane * 4 + j];
            vals[j] = v; s += v; sq += v * v;
        }
        #pragma unroll
        for (int off = 16; off > 0; off >>= 1) {
            s  += __shfl_xor(s,  off, 32);
            sq += __shfl_xor(sq, off, 32);
        }
        const float mu  = s * (1.0f / 128.0f);
        const float var = sq * (1.0f / 128.0f) - mu * mu;
        const float inv = rsqrtf(var + 0.001f);
        #pragma unroll
        for (int j = 0; j < 4; ++j) {
            int cc = lane * 4 + j;
            out[row * 128 + cc] = (vals[j] - mu) * inv * gamma[cc] + beta[cc];
        }
    }
}

// ---------------------------------------------------------------------------
extern "C" void kernel_launch(void* const* d_in, const int* in_sizes, int n_in,
                              void* d_out, int out_size, void* d_ws, size_t ws_size,
                              hipStream_t stream) {
    const float* H      = (const float*)d_in[0];
    const int*   indice = (const int*)  d_in[1];
    const float* W      = (const float*)d_in[2];
    const float* bfin   = (const float*)d_in[3];
    const float* gamma  = (const float*)d_in[4];
    const float* beta   = (const float*)d_in[5];
    float*       out    = (float*)d_out;

    float* partial = (float*)d_ws;                       // B*CHUNKS*128 floats
    float* emb     = partial + (size_t)B * CHUNKS * 128; // B*640 floats

    hipLaunchKernelGGL(k_reduce_partial, dim3(B * CHUNKS), dim3(256), 0, stream,
                       H, partial);
    hipLaunchKernelGGL(k_build_emb, dim3(B), dim3(128), 0, stream,
                       H, indice, partial, emb);
    hipLaunchKernelGGL(k_gemm_ln, dim3(1), dim3(1024), 0, stream,
                       emb, W, bfin, gamma, beta, out);
}